// SIDARTHEOde_12773232738687
// MI455X (gfx1250) — compile-verified
//
#include <hip/hip_runtime.h>

typedef __attribute__((ext_vector_type(2))) float v2f;
typedef __attribute__((ext_vector_type(8))) float v8f;

#define LEN_DATA 100000

__device__ __forceinline__ float rl(float v, int l) {
  return __builtin_bit_cast(float, __builtin_amdgcn_readlane(__builtin_bit_cast(int, v), l));
}

// One wave32. State (S,I,A,R,T,D,E,H) kept permanently in WMMA A-operand
// layout: a0 = K0|K2 = S|A, a1 = K1|K3 = I|R, a2 = K4|K6 = T|E,
// a3 = K5|K7 = D|H (lo-half | hi-half), plus a lane-uniform copy of S.
// RHS linear part + force dot-product (augmented column 8) via two chained
// V_WMMA_F32_16X16X4_F32. All RK combines are elementwise, so they are done
// directly in A-layout (4 FMAs + 1 uniform FMA each) -> zero operand prep
// in front of each WMMA.
struct Kv { float v0, v1, v2, v3, s; };

__global__ __launch_bounds__(32, 1)
void sidarthe_rk38_wmma(const float* __restrict__ rates,
                        const float* __restrict__ y0,
                        const float* __restrict__ D0,
                        const float* __restrict__ E0,
                        const float* __restrict__ H0,
                        float* __restrict__ out) {
  const int lane = threadIdx.x;
  const bool lohalf = lane < 16;

  // rate order: alpha beta gamma delta eps theta zeta eta mu nu tau lam kappa xi rho sigma
  const float al = rates[0],  be = rates[1],  ga = rates[2],  de = rates[3];
  const float ep = rates[4],  th = rates[5],  ze = rates[6],  et = rates[7];
  const float mu = rates[8],  nu = rates[9],  ta = rates[10], la = rates[11];
  const float ka = rates[12], xi = rates[13], rh = rates[14], si = rates[15];

  // Stage constant augmented matrix W[k][n] (k=input comp, n=output comp,
  // n==8 = force row) into B-operand lane layout via LDS (once).
  // Bv0: K0|K2, Bv1: K1|K3, Bv2: K4|K6, Bv3: K5|K7 (lo | hi half-wave).
  __shared__ float Wl[4 * 32];
  for (int i = lane; i < 128; i += 32) Wl[i] = 0.0f;
  __syncthreads();
  if (lane == 0) {
    // Bv0 hi: K=2 (A)
    Wl[0*32+16+2] = -(th + mu + ka);
    Wl[0*32+16+3] = th;
    Wl[0*32+16+4] = mu;
    Wl[0*32+16+7] = ka;
    Wl[0*32+16+8] = ga;
    // Bv1 lo: K=1 (I)
    Wl[1*32+ 0+1] = -(ep + ze + la);
    Wl[1*32+ 0+2] = ze;
    Wl[1*32+ 0+5] = ep;
    Wl[1*32+ 0+7] = la;
    Wl[1*32+ 0+8] = al;
    // Bv1 hi: K=3 (R)
    Wl[1*32+16+3] = -(nu + xi);
    Wl[1*32+16+4] = nu;
    Wl[1*32+16+7] = xi;
    Wl[1*32+16+8] = de;
    // Bv2 lo: K=4 (T)   (hi K=6 (E): all zero)
    Wl[2*32+ 0+4] = -(si + ta);
    Wl[2*32+ 0+6] = ta;
    Wl[2*32+ 0+7] = si;
    // Bv3 lo: K=5 (D)   (hi K=7 (H): all zero)
    Wl[3*32+ 0+3] = et;
    Wl[3*32+ 0+5] = -(et + rh);
    Wl[3*32+ 0+7] = rh;
    Wl[3*32+ 0+8] = be;
  }
  __syncthreads();
  v2f B01, B23;
  B01[0] = Wl[0*32 + lane];
  B01[1] = Wl[1*32 + lane];
  B23[0] = Wl[2*32 + lane];
  B23[1] = Wl[3*32 + lane];

  // Load initial state and move into A-layout.
  const float iS = y0[0], iI = y0[1], iA = y0[2], iR = y0[3], iT = y0[4];
  const float iD = D0[0], iE = E0[0], iH = H0[0];
  float a0 = lohalf ? iS : iA;
  float a1 = lohalf ? iI : iR;
  float a2 = lohalf ? iT : iE;
  float a3 = lohalf ? iD : iH;
  float S  = iS;                       // lane-uniform S component

  const bool wlane = (lane == 0) || (lane == 16);
  // Row 0 = initial state. lane0 writes comps (0,1),(4,5); lane16 (2,3),(6,7).
  if (wlane) {
    float* row = out + (lohalf ? 0 : 2);
    *(float2*)(row + 0) = make_float2(a0, a1);
    *(float2*)(row + 4) = make_float2(a2, a3);
  }

  auto rhs = [&](float x0, float x1, float x2, float x3, float xS) -> Kv {
    v2f A01, A23;
    A01[0] = x0; A01[1] = x1;
    A23[0] = x2; A23[1] = x3;
    v8f c = {};
    v8f d = __builtin_amdgcn_wmma_f32_16x16x4_f32(false, A01, false, B01,
                                                  (short)0, c, false, false);
    d = __builtin_amdgcn_wmma_f32_16x16x4_f32(false, A23, false, B23,
                                              (short)0, d, false, false);
    float lin = d[0];                  // lane n holds linear output n (n<=8)
    float force = rl(lin, 8);          // a*I + b*D + g*A + d*R
    float sf = xS * force;             // S * force  (uniform)
    float k0 = -sf;                    // dS
    float k1 = rl(lin, 1) + sf;        // dI
    float l2 = rl(lin, 2), l3 = rl(lin, 3);
    float l4 = rl(lin, 4), l5 = rl(lin, 5);
    float l6 = rl(lin, 6), l7 = rl(lin, 7);
    Kv k;
    k.v0 = lohalf ? k0 : l2;           // dS | dA
    k.v1 = lohalf ? k1 : l3;           // dI | dR
    k.v2 = lohalf ? l4 : l6;           // dT | dE
    k.v3 = lohalf ? l5 : l7;           // dD | dH
    k.s  = k0;
    return k;
  };

  const float c13 = 1.0f / 3.0f;       // dt == 1.0
  for (int t = 0; t < LEN_DATA; ++t) {
    Kv k1 = rhs(a0, a1, a2, a3, S);

    float b0 = __builtin_fmaf(c13, k1.v0, a0);
    float b1 = __builtin_fmaf(c13, k1.v1, a1);
    float b2 = __builtin_fmaf(c13, k1.v2, a2);
    float b3 = __builtin_fmaf(c13, k1.v3, a3);
    float bS = __builtin_fmaf(c13, k1.s,  S);
    Kv k2 = rhs(b0, b1, b2, b3, bS);

    float c0 = a0 + (k2.v0 - c13 * k1.v0);
    float c1 = a1 + (k2.v1 - c13 * k1.v1);
    float c2 = a2 + (k2.v2 - c13 * k1.v2);
    float c3 = a3 + (k2.v3 - c13 * k1.v3);
    float cS = S  + (k2.s  - c13 * k1.s);
    Kv k3 = rhs(c0, c1, c2, c3, cS);

    float d0 = a0 + (k1.v0 - k2.v0 + k3.v0);
    float d1 = a1 + (k1.v1 - k2.v1 + k3.v1);
    float d2 = a2 + (k1.v2 - k2.v2 + k3.v2);
    float d3 = a3 + (k1.v3 - k2.v3 + k3.v3);
    float dS = S  + (k1.s  - k2.s  + k3.s);
    Kv k4 = rhs(d0, d1, d2, d3, dS);

    a0 = __builtin_fmaf(0.125f, k1.v0 + 3.0f * (k2.v0 + k3.v0) + k4.v0, a0);
    a1 = __builtin_fmaf(0.125f, k1.v1 + 3.0f * (k2.v1 + k3.v1) + k4.v1, a1);
    a2 = __builtin_fmaf(0.125f, k1.v2 + 3.0f * (k2.v2 + k3.v2) + k4.v2, a2);
    a3 = __builtin_fmaf(0.125f, k1.v3 + 3.0f * (k2.v3 + k3.v3) + k4.v3, a3);
    S  = __builtin_fmaf(0.125f, k1.s  + 3.0f * (k2.s  + k3.s)  + k4.s,  S);

    if (wlane) {                       // stores off the dependency chain
      float* row = out + (size_t)(t + 1) * 8 + (lohalf ? 0 : 2);
      *(float2*)(row + 0) = make_float2(a0, a1);
      *(float2*)(row + 4) = make_float2(a2, a3);
    }
  }
}

extern "C" void kernel_launch(void* const* d_in, const int* in_sizes, int n_in,
                              void* d_out, int out_size, void* d_ws, size_t ws_size,
                              hipStream_t stream) {
  const float* rates = (const float*)d_in[0];
  const float* y0p   = (const float*)d_in[1];
  const float* D0p   = (const float*)d_in[2];
  const float* E0p   = (const float*)d_in[3];
  const float* H0p   = (const float*)d_in[4];
  float* out = (float*)d_out;
  sidarthe_rk38_wmma<<<1, 32, 0, stream>>>(rates, y0p, D0p, E0p, H0p, out);
  (void)in_sizes; (void)n_in; (void)out_size; (void)d_ws; (void)ws_size;
}